// SE_GATTransfomerOnlyDecoderNoEdges_23244363006450
// MI455X (gfx1250) — compile-verified
//
#include <hip/hip_runtime.h>
#include <math.h>

#define NN 50000
#define NE 1600000
#define ET (NE + NN)
#define NGRAPH 16
#define DMODEL 128

typedef float v2f __attribute__((ext_vector_type(2)));
typedef float v8f __attribute__((ext_vector_type(8)));

// order-preserving float->uint key (monotone): allows atomicMax(u32) for float max
__device__ __forceinline__ unsigned f2ord(float f) {
  unsigned b = __float_as_uint(f);
  return (b & 0x80000000u) ? ~b : (b | 0x80000000u);
}
__device__ __forceinline__ float ord2f(unsigned k) {
  unsigned b = (k & 0x80000000u) ? (k ^ 0x80000000u) : ~k;
  return __uint_as_float(b);
}
__device__ __forceinline__ float wave_sum(float v) {
#pragma unroll
  for (int o = 16; o > 0; o >>= 1) v += __shfl_xor(v, o, 32);
  return v;
}

__global__ void k_fill_f32(float* p, float v, int n) {
  int i = blockIdx.x * blockDim.x + threadIdx.x;
  if (i < n) p[i] = v;
}

// feature embed: [NN,8]@[8,4]+b
__global__ void k_feat(const float* __restrict__ x, const float* __restrict__ W,
                       const float* __restrict__ b, float* __restrict__ out) {
  int n = blockIdx.x * blockDim.x + threadIdx.x;
  if (n >= NN) return;
  const float* xr = x + n * 8;
#pragma unroll
  for (int o = 0; o < 4; ++o) {
    float s = b[o];
#pragma unroll
    for (int k = 0; k < 8; ++k) s += xr[k] * W[k * 4 + o];
    out[n * 4 + o] = s;
  }
}

// fp32 WMMA GEMM: C[M,Nc] = A[M,K] @ B[K,Nc] (+bias), optional activation.
// One wave per 16x16 output tile, K-loop in steps of 4 via V_WMMA_F32_16X16X4_F32.
// A-frag (16x4 f32): lane%16 = M row; lanes<16 hold K={0,1}, lanes>=16 K={2,3}.
// B-frag (4x16 f32): lane%16 = N col; same K split.  C/D: VGPR i -> M=i(+8), N=lane%16.
__global__ void k_gemm_wmma(const float* __restrict__ A, int lda,
                            const float* __restrict__ B, int ldb,
                            const float* __restrict__ bias,
                            float* __restrict__ C, int ldc,
                            int M, int K, int Nc, int act) {
  int wave = blockIdx.x * (blockDim.x >> 5) + (threadIdx.x >> 5);
  int tilesN = Nc >> 4;
  int tot = (M >> 4) * tilesN;
  if (wave >= tot) return;
  int tm = wave / tilesN, tn = wave - tm * tilesN;
  int lane = threadIdx.x & 31;
  int r = lane & 15, sub = lane >> 4;
  int m0 = tm << 4, n0 = tn << 4;
  float bv = bias ? bias[n0 + r] : 0.f;
  v8f acc = {bv, bv, bv, bv, bv, bv, bv, bv};
  const float* ap = A + (size_t)(m0 + r) * lda + (sub << 1);
  const float* bp = B + (size_t)(sub << 1) * ldb + n0 + r;
  for (int k = 0; k < K; k += 4) {
    v2f a, bb;
    a.x = ap[k];
    a.y = ap[k + 1];
    bb.x = bp[(size_t)k * ldb];
    bb.y = bp[(size_t)(k + 1) * ldb];
    acc = __builtin_amdgcn_wmma_f32_16x16x4_f32(false, a, false, bb, (short)0,
                                                acc, false, false);
  }
#pragma unroll
  for (int i = 0; i < 8; ++i) {
    float v = acc[i];
    if (act == 1) v = v > 0.f ? v : 0.f;
    else if (act == 2) v = v > 0.f ? v : 0.01f * v;
    C[(size_t)(m0 + i + (sub << 3)) * ldc + n0 + r] = v;
  }
}

// per-(node,head) attention logits a_s, a_d
__global__ void k_prep(const float* __restrict__ h, int C,
                       const float* __restrict__ asrc, const float* __restrict__ adst,
                       float* __restrict__ as_, float* __restrict__ ad_) {
  int i = blockIdx.x * blockDim.x + threadIdx.x;
  if (i >= NN * 4) return;
  int n = i >> 2, hd = i & 3;
  int HC = C << 2;
  const float* hp = h + (size_t)n * HC + hd * C;
  const float* s = asrc + hd * C;
  const float* d = adst + hd * C;
  float vs = 0.f, vd = 0.f;
  for (int c = 0; c < C; ++c) { vs += hp[c] * s[c]; vd += hp[c] * d[c]; }
  as_[i] = vs;
  ad_[i] = vd;
}

__global__ void k_edge_max(const int* __restrict__ ei, const float* __restrict__ as_,
                           const float* __restrict__ ad_, unsigned* __restrict__ m) {
  int e = blockIdx.x * blockDim.x + threadIdx.x;
  if (e >= ET) return;
  int s, d;
  if (e < NE) { s = ei[e]; d = ei[NE + e]; } else { s = d = e - NE; }
#pragma unroll
  for (int hd = 0; hd < 4; ++hd) {
    float v = as_[s * 4 + hd] + ad_[d * 4 + hd];
    v = v > 0.f ? v : 0.2f * v;  // leaky_relu(0.2)
    atomicMax(&m[d * 4 + hd], f2ord(v));
  }
}

__global__ void k_edge_exp(const int* __restrict__ ei, const float* __restrict__ as_,
                           const float* __restrict__ ad_, const unsigned* __restrict__ m,
                           float* __restrict__ denom, float* __restrict__ wedge) {
  int e = blockIdx.x * blockDim.x + threadIdx.x;
  if (e >= ET) return;
  int s, d;
  if (e < NE) { s = ei[e]; d = ei[NE + e]; } else { s = d = e - NE; }
#pragma unroll
  for (int hd = 0; hd < 4; ++hd) {
    float v = as_[s * 4 + hd] + ad_[d * 4 + hd];
    v = v > 0.f ? v : 0.2f * v;
    float w = expf(v - ord2f(m[d * 4 + hd]));
    wedge[(size_t)e * 4 + hd] = w;
    atomicAdd(&denom[d * 4 + hd], w);
  }
}

__global__ void k_bias_init(float* __restrict__ out, const float* __restrict__ b,
                            int mask, int total) {
  int i = blockIdx.x * blockDim.x + threadIdx.x;
  if (i < total) out[i] = b[i & mask];
}

// one wave per edge: gather h[src], scale by alpha(head), atomic-add into out[dst]
__global__ void k_edge_scatter(const int* __restrict__ ei, const float* __restrict__ h,
                               int HC, int logC, const float* __restrict__ wedge,
                               const float* __restrict__ denom, float* __restrict__ out) {
  int e = blockIdx.x * (blockDim.x >> 5) + (threadIdx.x >> 5);
  if (e >= ET) return;
  int lane = threadIdx.x & 31;
  int s, d;
  if (e < NE) { s = ei[e]; d = ei[NE + e]; } else { s = d = e - NE; }
  for (int c = lane; c < HC; c += 32) {
    int hd = c >> logC;
    float alpha = wedge[(size_t)e * 4 + hd] / (denom[d * 4 + hd] + 1e-16f);
    atomicAdd(&out[(size_t)d * HC + c], h[(size_t)s * HC + c] * alpha);
  }
}

__global__ void k_lrelu(float* p, int n, float slope) {
  int i = blockIdx.x * blockDim.x + threadIdx.x;
  if (i < n) { float v = p[i]; p[i] = v > 0.f ? v : slope * v; }
}

// out = LayerNorm(xa + xb), one wave per node, 128 features
__global__ void k_ln_add(const float* __restrict__ xa, const float* __restrict__ xb,
                         const float* __restrict__ g, const float* __restrict__ b,
                         float* __restrict__ out) {
  int n = blockIdx.x * (blockDim.x >> 5) + (threadIdx.x >> 5);
  if (n >= NN) return;
  int lane = threadIdx.x & 31;
  float v[4];
  float s = 0.f;
#pragma unroll
  for (int j = 0; j < 4; ++j) {
    int c = lane + (j << 5);
    v[j] = xa[(size_t)n * DMODEL + c] + xb[(size_t)n * DMODEL + c];
    s += v[j];
  }
  s = wave_sum(s);
  float mu = s * (1.f / DMODEL);
  float q = 0.f;
#pragma unroll
  for (int j = 0; j < 4; ++j) { float t = v[j] - mu; q += t * t; }
  q = wave_sum(q);
  float inv = rsqrtf(q * (1.f / DMODEL) + 1e-5f);
#pragma unroll
  for (int j = 0; j < 4; ++j) {
    int c = lane + (j << 5);
    out[(size_t)n * DMODEL + c] = (v[j] - mu) * inv * g[c] + b[c];
  }
}

__global__ void k_gate_dot(const float* __restrict__ gh, const float* __restrict__ W2,
                           const float* __restrict__ b2, float* __restrict__ gate) {
  int n = blockIdx.x * blockDim.x + threadIdx.x;
  if (n >= NN) return;
  float s = b2[0];
#pragma unroll
  for (int k = 0; k < 32; ++k) s += gh[(size_t)n * 32 + k] * W2[k];
  gate[n] = s;
}

__global__ void k_seg_max(const float* __restrict__ gate, const int* __restrict__ batch,
                          unsigned* __restrict__ gm) {
  int n = blockIdx.x * blockDim.x + threadIdx.x;
  if (n >= NN) return;
  atomicMax(&gm[batch[n]], f2ord(gate[n]));
}

__global__ void k_seg_expsum(float* __restrict__ gate, const int* __restrict__ batch,
                             const unsigned* __restrict__ gm, float* __restrict__ gs) {
  int n = blockIdx.x * blockDim.x + threadIdx.x;
  if (n >= NN) return;
  int b = batch[n];
  float g = expf(gate[n] - ord2f(gm[b]));
  gate[n] = g;
  atomicAdd(&gs[b], g);
}

__global__ void k_pool(const float* __restrict__ x, const float* __restrict__ ge,
                       const float* __restrict__ gs, const int* __restrict__ batch,
                       float* __restrict__ pooled) {
  int n = blockIdx.x * (blockDim.x >> 5) + (threadIdx.x >> 5);
  if (n >= NN) return;
  int lane = threadIdx.x & 31;
  int b = batch[n];
  float wgt = ge[n] / (gs[b] + 1e-16f);
#pragma unroll
  for (int j = 0; j < 4; ++j) {
    int c = lane + (j << 5);
    atomicAdd(&pooled[(size_t)b * DMODEL + c], wgt * x[(size_t)n * DMODEL + c]);
  }
}

extern "C" void kernel_launch(void* const* d_in, const int* in_sizes, int n_in,
                              void* d_out, int out_size, void* d_ws, size_t ws_size,
                              hipStream_t stream) {
  auto F = [&](int i) { return (const float*)d_in[i]; };
  const float* x = F(0);
  const int* ei = (const int*)d_in[1];
  const int* batch = (const int*)d_in[2];
  // params in jax pytree (recursively key-sorted) flatten order
  const float* dW1 = F(3);   // dec[0].W1 [128,64]
  const float* dW2 = F(4);   // dec[0].W2 [64,128]
  const float* db1 = F(5);
  const float* db2 = F(6);
  const float* caWo = F(8);  // ca: Wk(7) Wo(8) Wq(9) Wv(10) bk(11) bo(12) bq(13) bv(14)
  const float* caWv = F(10);
  const float* cabo = F(12);
  const float* cabv = F(14);
  const float* ln1b = F(15); const float* ln1g = F(16);
  const float* ln2b = F(17); const float* ln2g = F(18);
  const float* ln3b = F(19); const float* ln3g = F(20);
  const float* saWo = F(22); // sa: Wk(21) Wo(22) Wq(23) Wv(24) bk(25) bo(26) bq(27) bv(28)
  const float* saWv = F(24);
  const float* sabo = F(26);
  const float* sabv = F(28);
  const float* fcW = F(29); const float* fcb = F(30);
  const float* featW = F(31); const float* featb = F(32);
  const float* g0W = F(33); const float* g0ad = F(34); const float* g0as = F(35); const float* g0b = F(36);
  const float* g1W = F(37); const float* g1ad = F(38); const float* g1as = F(39); const float* g1b = F(40);
  const float* g2W = F(41); const float* g2ad = F(42); const float* g2as = F(43); const float* g2b = F(44);
  const float* gtW1 = F(45); const float* gtW2 = F(46); const float* gtb1 = F(47); const float* gtb2 = F(48);
  const float* pW = F(49); const float* pad_ = F(50); const float* pas_ = F(51); const float* pb = F(52);

  float* w = (float*)d_ws;
  size_t off = 0;
  auto alloc = [&](size_t ne) { float* p = w + off; off += ne; return p; };
  float* xe   = alloc((size_t)NN * 4);
  float* as_  = alloc((size_t)NN * 4);
  float* ad_  = alloc((size_t)NN * 4);
  float* mbuf = alloc((size_t)NN * 4);   // unsigned keys
  float* dnm  = alloc((size_t)NN * 4);
  float* wed  = alloc((size_t)ET * 4);
  float* bufH = alloc((size_t)NN * 128);
  float* bufX = alloc((size_t)NN * 128);
  float* bufY = alloc((size_t)NN * 128);
  float* bufZ = alloc((size_t)NN * 128);
  float* bufV = alloc((size_t)NN * 128);
  float* gate = alloc((size_t)NN);
  float* gm   = alloc(16);
  float* gs   = alloc(16);
  float* pool = alloc((size_t)NGRAPH * 128);

  auto gemm = [&](const float* A, int lda, const float* B, int ldb, const float* bias,
                  float* C, int ldc, int M, int K, int Nc, int act) {
    int tiles = (M >> 4) * (Nc >> 4);
    int blocks = (tiles + 3) >> 2;
    k_gemm_wmma<<<blocks, 128, 0, stream>>>(A, lda, B, ldb, bias, C, ldc, M, K, Nc, act);
  };
  auto fill = [&](float* p, float v, int n) {
    k_fill_f32<<<(n + 255) / 256, 256, 0, stream>>>(p, v, n);
  };
  auto gatL = [&](const float* xin, int Kin, const float* Wp, const float* adv,
                  const float* asv, const float* bp, int C, float* outb, int act) {
    int HC = C << 2;
    gemm(xin, Kin, Wp, HC, nullptr, bufH, HC, NN, Kin, HC, 0);
    k_prep<<<(NN * 4 + 255) / 256, 256, 0, stream>>>(bufH, C, asv, adv, as_, ad_);
    fill(mbuf, 0.f, NN * 4);  // key 0 == -inf in ordered-uint space
    fill(dnm, 0.f, NN * 4);
    k_edge_max<<<(ET + 255) / 256, 256, 0, stream>>>(ei, as_, ad_, (unsigned*)mbuf);
    k_edge_exp<<<(ET + 255) / 256, 256, 0, stream>>>(ei, as_, ad_, (const unsigned*)mbuf, dnm, wed);
    k_bias_init<<<(NN * HC + 255) / 256, 256, 0, stream>>>(outb, bp, HC - 1, NN * HC);
    k_edge_scatter<<<(ET + 7) / 8, 256, 0, stream>>>(ei, bufH, HC, C == 16 ? 4 : 5, wed, dnm, outb);
    if (act) k_lrelu<<<(NN * HC + 255) / 256, 256, 0, stream>>>(outb, NN * HC, 0.01f);
  };

  // ---- GAT encoder ----
  k_feat<<<(NN + 255) / 256, 256, 0, stream>>>(x, featW, featb, xe);
  gatL(xe, 4, g0W, g0ad, g0as, g0b, 16, bufX, 0);
  gatL(bufX, 64, g1W, g1ad, g1as, g1b, 16, bufY, 1);
  gatL(bufY, 64, g2W, g2ad, g2as, g2b, 16, bufX, 1);
  gatL(bufX, 64, pW, pad_, pas_, pb, 32, bufY, 0);  // P = bufY [NN,128]

  // ---- decoder: S=T=1 => attention == V path (q,k mathematically irrelevant) ----
  gemm(bufY, 128, saWv, 128, sabv, bufZ, 128, NN, 128, 128, 0);
  gemm(bufZ, 128, saWo, 128, sabo, bufV, 128, NN, 128, 128, 0);
  k_ln_add<<<NN / 8, 256, 0, stream>>>(bufY, bufV, ln1g, ln1b, bufZ);  // x1
  gemm(bufY, 128, caWv, 128, cabv, bufV, 128, NN, 128, 128, 0);
  gemm(bufV, 128, caWo, 128, cabo, bufX, 128, NN, 128, 128, 0);
  k_ln_add<<<NN / 8, 256, 0, stream>>>(bufZ, bufX, ln2g, ln2b, bufV);  // x2
  gemm(bufV, 128, dW1, 64, db1, bufX, 64, NN, 128, 64, 1);             // relu FFN
  gemm(bufX, 64, dW2, 128, db2, bufZ, 128, NN, 64, 128, 0);
  k_ln_add<<<NN / 8, 256, 0, stream>>>(bufV, bufZ, ln3g, ln3b, bufX);  // x3

  // ---- global attention pooling + output FC ----
  gemm(bufX, 128, gtW1, 32, gtb1, bufZ, 32, NN, 128, 32, 1);
  k_gate_dot<<<(NN + 255) / 256, 256, 0, stream>>>(bufZ, gtW2, gtb2, gate);
  fill(gm, 0.f, 16);
  fill(gs, 0.f, 16);
  fill(pool, 0.f, NGRAPH * 128);
  k_seg_max<<<(NN + 255) / 256, 256, 0, stream>>>(gate, batch, (unsigned*)gm);
  k_seg_expsum<<<(NN + 255) / 256, 256, 0, stream>>>(gate, batch, (const unsigned*)gm, gs);
  k_pool<<<NN / 8, 256, 0, stream>>>(bufX, gate, gs, batch, pool);
  gemm(pool, 128, fcW, 128, fcb, (float*)d_out, 128, 16, 128, 128, 0);

  (void)in_sizes; (void)n_in; (void)out_size; (void)ws_size;
}